// CapsuleLayer_56135222558899
// MI455X (gfx1250) — compile-verified
//
#include <hip/hip_runtime.h>
#include <hip/hip_bf16.h>
#include <math.h>
#include <stdint.h>

// ---------------------------------------------------------------------------
// CapsNet dynamic routing, fully fused for MI455X (gfx1250, wave32, WMMA).
// u_hat is NEVER materialized in HBM: it is recomputed from x/W (L2-resident,
// ~40 MB) via V_WMMA_F32_16X16X4_F32 each routing pass. LDS staging uses the
// CDNA5 async global->LDS DMA (ASYNCcnt), double-buffered so the DMA for
// tile i+1 overlaps compute on tile i (async loads retire in order, so
// s_wait_asynccnt NEWK == "previous buffer complete").
// ---------------------------------------------------------------------------

typedef float v2f __attribute__((ext_vector_type(2)));
typedef float v8f __attribute__((ext_vector_type(8)));

#define NB 64      // batch
#define NI 1152    // in-capsules
#define NO 64      // out-capsules
#define ND 8       // in dim (K of GEMM)
#define NC 16      // out dim (N of GEMM)
#define I_PER_BLK 8
#define THREADS 1024   // 32 waves of 32

// D = A(16x4 f32) * B(4x16 f32) + C(16x16 f32)
__device__ __forceinline__ v8f wmma4(v2f a, v2f b, v8f c) {
    return __builtin_amdgcn_wmma_f32_16x16x4_f32(
        false, a, false, b, (short)0, c, false, false);
}

// ---- CDNA5 async global->LDS DMA (tracked by ASYNCcnt) --------------------
__device__ __forceinline__ uint32_t lds_off(const void* p) {
    return (uint32_t)(uintptr_t)p;   // flat LDS pointer: low 32 bits = LDS addr
}
__device__ __forceinline__ void async_ld_b128(uint32_t lds, const void* g) {
    asm volatile("global_load_async_to_lds_b128 %0, %1, off"
                 :: "v"(lds), "v"(g) : "memory");
}
__device__ __forceinline__ void async_ld_b32(uint32_t lds, const void* g) {
    asm volatile("global_load_async_to_lds_b32 %0, %1, off"
                 :: "v"(lds), "v"(g) : "memory");
}
template <int N>
__device__ __forceinline__ void wait_async() {
    asm volatile("s_wait_asynccnt %0" :: "i"(N) : "memory");
}

// ---------------------------------------------------------------------------
// Routing kernel. Grid: (NI/I_PER_BLK)*2 blocks; block = (i-chunk, b-half).
// ITER 0: c = 1/64 uniform, accumulate s only.
// ITER 1: agreement with v0 -> b_logits (write) -> softmax -> s.
// ITER 2: b_logits(DMA read) + agreement with v1 -> softmax -> s.
// ---------------------------------------------------------------------------
template <int ITER>
__global__ __launch_bounds__(THREADS) void route_kernel(
    const float* __restrict__ x,      // [b][i][d]
    const float* __restrict__ W,      // [i][o][d][c]
    const float* __restrict__ v_in,   // [b][o][c]
    float*       __restrict__ blog,   // [b][i][o]
    float*       __restrict__ s_out)  // [b][o][c]
{
    __shared__ float W_s[2][NO * ND * NC];            // 64 KB, natural [o][d][c]
    __shared__ float x_s[2][32 * 10];                 // 2.5 KB, [b][d pad 10]
    __shared__ float b_row[(ITER == 2) ? 2 : 1][32 * NO];  // 8/16 KB

    const int tid   = threadIdx.x;
    const int wave  = tid >> 5;
    const int lane  = tid & 31;
    const int half  = lane >> 4;
    const int lm    = lane & 15;
    const int ic    = blockIdx.x >> 1;
    const int bh    = blockIdx.x & 1;
    const int bbase = bh * 32;
    const int i0    = ic * I_PER_BLK;
    const int d0    = 2 * half;

    // asyncs issued per wave per stage (wave-uniform by construction)
    constexpr int NEWK = (ITER == 2) ? 4 : 3;

    // issue the async-DMA stage for tile i into buffer `buf`
    auto issue = [&](int i, int buf) {
        const char* Wi = (const char*)(W + (size_t)i * NO * ND * NC);
        uint32_t wl = lds_off(W_s[buf]);
        async_ld_b128(wl + tid * 16u,             Wi + (size_t)tid * 16);
        async_ld_b128(wl + (tid + THREADS) * 16u, Wi + (size_t)(tid + THREADS) * 16);
        {   // x slab; waves 8..31 duplicate waves 0..7 (keeps NEWK uniform)
            int e = tid & 255, b = e >> 3, d = e & 7;
            async_ld_b32(lds_off(&x_s[buf][b * 10 + d]),
                         &x[((size_t)(bbase + b) * NI + i) * ND + d]);
        }
        if (ITER == 2) {   // b_logits row; waves 16..31 duplicate 0..15
            int ch = tid & 511, b = ch >> 4, o0 = (ch & 15) * 4;
            async_ld_b128(lds_off(b_row[buf]) + ch * 16u,
                          &blog[((size_t)(bbase + b) * NI + i) * NO + o0]);
        }
    };

    // 4 persistent accumulators per wave: units = 64 o * 2 b-tiles = 128
    v8f acc[4];
#pragma unroll
    for (int q = 0; q < 4; ++q) acc[q] = {};

    issue(i0, 0);   // prologue: stage first tile

    for (int ii = 0; ii < I_PER_BLK; ++ii) {
        const int i   = i0 + ii;
        const int cur = ii & 1;

        if (ii + 1 < I_PER_BLK) {
            issue(i + 1, cur ^ 1);                     // overlap DMA w/ compute
            if (ii + 2 < I_PER_BLK)                    // warm L2 one more ahead
                __builtin_prefetch((const char*)(W + (size_t)(i + 2) * NO * ND * NC)
                                       + tid * 32, 0, 1);
            wait_async<NEWK>();   // loads retire in order: prev buffer ready
        } else {
            wait_async<0>();
        }
        __syncthreads();

        const float* Ws = W_s[cur];
        const float* xs = x_s[cur];
        float*       br = b_row[(ITER == 2) ? cur : 0];

        if (ITER == 1) {          // zero logits row (prev value is 0)
            for (int e = tid; e < 32 * NO; e += THREADS) br[e] = 0.f;
            __syncthreads();
        }

        if (ITER > 0) {
            // ---- pass 1: u_hat (fresh) -> agreement -> b_row -------------
#pragma unroll
            for (int q = 0; q < 4; ++q) {
                const int unit = wave * 4 + q;   // 0..127
                const int o    = unit >> 1;
                const int bl0  = (unit & 1) * 16;

                const float* xr = &xs[(bl0 + lm) * 10];
                v2f a0 = { xr[d0],     xr[d0 + 1] };
                v2f a1 = { xr[4 + d0], xr[5 + d0] };
                const float* wr = &Ws[o * ND * NC + lm];
                v2f b0 = { wr[(d0    ) * NC], wr[(d0 + 1) * NC] };
                v2f b1 = { wr[(d0 + 4) * NC], wr[(d0 + 5) * NC] };

                v8f z = {};
                v8f u = wmma4(a0, b0, z);       // K = 0..3
                u     = wmma4(a1, b1, u);       // K = 4..7  -> u_hat tile
#pragma unroll
                for (int j = 0; j < 8; ++j) {
                    const int bl = bl0 + j + 8 * half;
                    const int b  = bbase + bl;
                    float t = u[j] * v_in[((size_t)b * NO + o) * NC + lm];
                    t += __shfl_xor(t, 1, 32);   // reduce over c (16 lanes)
                    t += __shfl_xor(t, 2, 32);
                    t += __shfl_xor(t, 4, 32);
                    t += __shfl_xor(t, 8, 32);
                    if (lm == 0) br[bl * NO + o] += t;   // exclusive writer
                }
            }
            __syncthreads();

            if (ITER == 1) {   // persist b_logits (tiny b128 round trip)
                if (tid < 512) {
                    int b = tid >> 4, o0 = (tid & 15) * 4;
                    float4 t = *(const float4*)&br[tid * 4];
                    *(float4*)&blog[((size_t)(bbase + b) * NI + i) * NO + o0] = t;
                }
                __syncthreads();
            }

            // ---- softmax over o, in place (b_row becomes c) --------------
            if (tid < 32) {
                float* row = &br[tid * NO];
                float m = -3.4e38f;
                for (int o = 0; o < NO; ++o) m = fmaxf(m, row[o]);
                float sum = 0.f;
                for (int o = 0; o < NO; ++o) {
                    float e2 = __expf(row[o] - m);
                    row[o] = e2;
                    sum += e2;
                }
                float inv = 1.f / sum;
                for (int o = 0; o < NO; ++o) row[o] *= inv;
            }
            __syncthreads();
        }

        // ---- pass 2: c-weighted u_hat accumulated in WMMA C operand ------
#pragma unroll
        for (int q = 0; q < 4; ++q) {
            const int unit = wave * 4 + q;
            const int o    = unit >> 1;
            const int bl0  = (unit & 1) * 16;

            const float scale =
                (ITER == 0) ? (1.0f / 64.0f) : br[(bl0 + lm) * NO + o];
            const float* xr = &xs[(bl0 + lm) * 10];
            // scaling A rows by c[b,o] == scaling D rows (s contribution)
            v2f a0 = { scale * xr[d0],     scale * xr[d0 + 1] };
            v2f a1 = { scale * xr[4 + d0], scale * xr[5 + d0] };
            const float* wr = &Ws[o * ND * NC + lm];
            v2f b0 = { wr[(d0    ) * NC], wr[(d0 + 1) * NC] };
            v2f b1 = { wr[(d0 + 4) * NC], wr[(d0 + 5) * NC] };

            acc[q] = wmma4(a0, b0, acc[q]);
            acc[q] = wmma4(a1, b1, acc[q]);
        }
        __syncthreads();   // buffer reuse safety for the next DMA issue
    }

    // ---- flush partial s (summed over this block's 8 i's) ----------------
#pragma unroll
    for (int q = 0; q < 4; ++q) {
        const int unit = wave * 4 + q;
        const int o    = unit >> 1;
        const int bl0  = (unit & 1) * 16;
#pragma unroll
        for (int j = 0; j < 8; ++j) {
            const int b = bbase + bl0 + j + 8 * half;
            atomicAdd(&s_out[((size_t)b * NO + o) * NC + lm], acc[q][j]);
        }
    }
}

__global__ void zero_kernel(float* __restrict__ p, int n) {
    int idx = blockIdx.x * blockDim.x + threadIdx.x;
    if (idx < n) p[idx] = 0.f;
}

__global__ void squash_kernel(const float* __restrict__ s,
                              float* __restrict__ v) {
    int row = blockIdx.x * blockDim.x + threadIdx.x;   // [b*o] rows, 4096
    if (row >= NB * NO) return;
    const float* sp = s + (size_t)row * NC;
    float vals[NC];
    float sq = 0.f;
#pragma unroll
    for (int c = 0; c < NC; ++c) { vals[c] = sp[c]; sq += vals[c] * vals[c]; }
    float scale = (sq / (1.f + sq)) / (sqrtf(sq) + 1e-8f);
#pragma unroll
    for (int c = 0; c < NC; ++c) v[(size_t)row * NC + c] = scale * vals[c];
}

extern "C" void kernel_launch(void* const* d_in, const int* in_sizes, int n_in,
                              void* d_out, int out_size, void* d_ws, size_t ws_size,
                              hipStream_t stream) {
    const float* x = (const float*)d_in[0];   // [64,1152,8]
    const float* W = (const float*)d_in[1];   // [1152,64,8,16]
    float* out = (float*)d_out;               // [64,64,16]

    float* ws   = (float*)d_ws;
    float* s    = ws;                         // 65536 floats
    float* v    = ws + 65536;                 // 65536 floats
    float* blog = ws + 131072;                // 64*1152*64 floats (~18.9 MB)

    const dim3 rgrid((NI / I_PER_BLK) * 2);   // 288 blocks
    const dim3 rblk(THREADS);
    const int  svn = NB * NO * NC;            // 65536

    // iter 0: c uniform
    zero_kernel<<<(svn + 255) / 256, 256, 0, stream>>>(s, svn);
    route_kernel<0><<<rgrid, rblk, 0, stream>>>(x, W, v, blog, s);
    squash_kernel<<<16, 256, 0, stream>>>(s, v);

    // iter 1: agreement(v0) -> b_logits -> softmax
    zero_kernel<<<(svn + 255) / 256, 256, 0, stream>>>(s, svn);
    route_kernel<1><<<rgrid, rblk, 0, stream>>>(x, W, v, blog, s);
    squash_kernel<<<16, 256, 0, stream>>>(s, v);

    // iter 2: b_logits + agreement(v1) -> softmax -> final v
    zero_kernel<<<(svn + 255) / 256, 256, 0, stream>>>(s, svn);
    route_kernel<2><<<rgrid, rblk, 0, stream>>>(x, W, v, blog, s);
    squash_kernel<<<16, 256, 0, stream>>>(s, out);
}